// ConvAttention_mask_19327352832684
// MI455X (gfx1250) — compile-verified
//
#include <hip/hip_runtime.h>
#include <math.h>

#define BATCH 4
#define IMG 32
#define NPOS 1024
#define DIM 256
#define HEADS 8
#define DHEAD 64
#define INNER 512
#define MROWS (BATCH * NPOS) /* 4096 */
#define SCALE 0.125f         /* DHEAD^-0.5 */

typedef __attribute__((ext_vector_type(2))) float v2f;
typedef __attribute__((ext_vector_type(8))) float v8f;

// fp32 WMMA: D(16x16) = A(16x4) * B(4x16) + C. wave32, frags are 2 VGPRs.
__device__ __forceinline__ v8f wmma4(v2f a, v2f b, v8f c) {
  return __builtin_amdgcn_wmma_f32_16x16x4_f32(false, a, false, b, (short)0, c,
                                               false, false);
}

// Async global->LDS 16B copy (ASYNCcnt-tracked, GV addressing mode).
__device__ __forceinline__ void async_copy16(void* lds_dst, const void* gsrc) {
  unsigned la = (unsigned)(size_t)lds_dst;  // LDS aperture: offset = addr[31:0]
  asm volatile("global_load_async_to_lds_b128 %0, %1, off" ::"v"(la), "v"(gsrc)
               : "memory");
}
__device__ __forceinline__ void wait_async0() {
  asm volatile("s_wait_asynccnt 0" ::: "memory");
}

// ---------------------------------------------------------------- pool ----
// x: [B, N, DIM] -> xattn: [B, 2, N] (mean over C, max over C)
__global__ void pool_kernel(const float* __restrict__ x,
                            float* __restrict__ xattn) {
  int idx = blockIdx.x * blockDim.x + threadIdx.x;  // b*1024 + pos
  const float* row = x + (size_t)idx * DIM;
  float s = 0.f, mx = -INFINITY;
  for (int c = 0; c < DIM; ++c) {
    float v = row[c];
    s += v;
    mx = fmaxf(mx, v);
  }
  int b = idx >> 10, pos = idx & 1023;
  xattn[(size_t)b * 2 * NPOS + pos] = s * (1.0f / DIM);
  xattn[(size_t)b * 2 * NPOS + NPOS + pos] = mx;
}

// ------------------------------------------------------------ mask conv ---
// qm/km[b][h][pos][d] from 3x3 conv over xattn [b,2,32,32]; o = h*2 + d
__global__ void maskproj_kernel(const float* __restrict__ xattn,
                                const float* __restrict__ qw,
                                const float* __restrict__ qb,
                                const float* __restrict__ kw,
                                const float* __restrict__ kb,
                                float* __restrict__ qm, float* __restrict__ km) {
  int t = blockIdx.x * blockDim.x + threadIdx.x;
  int sel = t >> 16;  // 0 = q, 1 = k
  int id = t & 65535;
  int pos = id & 1023;
  int o = (id >> 10) & 15;
  int b = id >> 14;
  const float* w = sel ? kw : qw;  // [16][2][3][3]
  const float* bb = sel ? kb : qb;
  int y = pos >> 5, xx = pos & 31;
  float acc = bb[o];
  for (int ci = 0; ci < 2; ++ci)
    for (int ky = 0; ky < 3; ++ky) {
      int py = y + ky - 1;
      if (py < 0 || py >= IMG) continue;
      for (int kx = 0; kx < 3; ++kx) {
        int px = xx + kx - 1;
        if (px < 0 || px >= IMG) continue;
        acc += w[(o * 2 + ci) * 9 + ky * 3 + kx] *
               xattn[((size_t)b * 2 + ci) * NPOS + py * IMG + px];
      }
    }
  int h = o >> 1, d = o & 1;
  float* dst = sel ? km : qm;
  dst[(((size_t)b * HEADS + h) * NPOS + pos) * 2 + d] = acc;
}

// -------------------------------------------------------------- top-4 -----
// per (b,h,i): top-4 j of rank-2 logit qm_i . km_j  (softmax is monotone,
// so top-k of softmax(dots_mask) == top-k of the raw logits).
__global__ void topk_kernel(const float* __restrict__ qm,
                            const float* __restrict__ km,
                            int* __restrict__ midx) {
  int wave = threadIdx.x >> 5;
  int lane = threadIdx.x & 31;
  int row = blockIdx.x * 4 + wave;  // global over b*h*n
  int bh = row >> 10;
  int i = row & 1023;
  const float* qv = qm + ((size_t)bh * NPOS + i) * 2;
  float a0 = qv[0], a1 = qv[1];
  const float* kb = km + (size_t)bh * NPOS * 2;
  float tv[4] = {-INFINITY, -INFINITY, -INFINITY, -INFINITY};
  int ti[4] = {-1, -1, -1, -1};
  for (int j = lane; j < NPOS; j += 32) {
    float v = a0 * kb[2 * j] + a1 * kb[2 * j + 1];
    if (v > tv[3]) {
      int p = 3;
      while (p > 0 && v > tv[p - 1]) {
        tv[p] = tv[p - 1];
        ti[p] = ti[p - 1];
        --p;
      }
      tv[p] = v;
      ti[p] = j;
    }
  }
  for (int off = 16; off >= 1; off >>= 1) {
    float ov[4];
    int oi[4];
#pragma unroll
    for (int s = 0; s < 4; ++s) {
      ov[s] = __shfl_xor(tv[s], off);
      oi[s] = __shfl_xor(ti[s], off);
    }
    float mv[4];
    int mi[4];
    int pa = 0, pb = 0;
#pragma unroll
    for (int s = 0; s < 4; ++s) {
      if (pb >= 4 || (pa < 4 && tv[pa] >= ov[pb])) {
        mv[s] = tv[pa];
        mi[s] = ti[pa];
        ++pa;
      } else {
        mv[s] = ov[pb];
        mi[s] = oi[pb];
        ++pb;
      }
    }
#pragma unroll
    for (int s = 0; s < 4; ++s) {
      tv[s] = mv[s];
      ti[s] = mi[s];
    }
  }
  if (lane == 0) {
    *(int4*)(midx + (size_t)row * 4) = make_int4(ti[0], ti[1], ti[2], ti[3]);
  }
}

// ------------------------------------------- depthwise + BN + GELU --------
// block = channel. 3x3 depthwise conv for all 4096 (b,pos), training-mode
// batch mean/var, exact-erf GELU; writes act[row=(b*1024+pos)][c].
__global__ void sepconv_kernel(const float* __restrict__ x,
                               const float* __restrict__ dw,
                               const float* __restrict__ db,
                               const float* __restrict__ g,
                               const float* __restrict__ beta,
                               float* __restrict__ act) {
  int c = blockIdx.x;
  int t = threadIdx.x;  // 0..255
  float w[9];
#pragma unroll
  for (int i = 0; i < 9; ++i) w[i] = dw[c * 9 + i];
  float bias = db[c];
  float yv[16];
  float s = 0.f, sq = 0.f;
#pragma unroll 4
  for (int k = 0; k < 16; ++k) {
    int idx = t + 256 * k;  // b*1024 + pos
    int b = idx >> 10, pos = idx & 1023;
    int yy = pos >> 5, xx = pos & 31;
    const float* xb = x + (size_t)b * NPOS * DIM;
    float acc = bias;
    for (int ky = 0; ky < 3; ++ky) {
      int py = yy + ky - 1;
      if (py < 0 || py >= IMG) continue;
      for (int kx = 0; kx < 3; ++kx) {
        int px = xx + kx - 1;
        if (px < 0 || px >= IMG) continue;
        acc += w[ky * 3 + kx] * xb[(size_t)(py * IMG + px) * DIM + c];
      }
    }
    yv[k] = acc;
    s += acc;
    sq += acc * acc;
  }
  __shared__ float ssum[256], ssq[256];
  ssum[t] = s;
  ssq[t] = sq;
  __syncthreads();
  for (int off = 128; off > 0; off >>= 1) {
    if (t < off) {
      ssum[t] += ssum[t + off];
      ssq[t] += ssq[t + off];
    }
    __syncthreads();
  }
  float mu = ssum[0] * (1.0f / MROWS);
  float var = ssq[0] * (1.0f / MROWS) - mu * mu;
  float rstd = rsqrtf(var + 1e-5f);
  float gc = g[c], bc = beta[c];
#pragma unroll 4
  for (int k = 0; k < 16; ++k) {
    int idx = t + 256 * k;
    float hv = (yv[k] - mu) * rstd * gc + bc;
    float ge = 0.5f * hv * (1.0f + erff(hv * 0.70710678118654752f));
    act[(size_t)idx * DIM + c] = ge;
  }
}

// -------------------------------- async double-buffered WMMA GEMM ---------
// C = A(MxK) * W(NxK)^T + bias. 4 waves per block, 32x32 C-tile (2x2 wave
// tiles). K consumed in 32-wide chunks staged in LDS via
// global_load_async_to_lds_b128, double buffered with s_wait_asynccnt.
// LDS row stride 36 floats (144B): 16B-aligned rows, 36r mod 64 distinct
// for r in [0,16) -> conflict-free ds_load_b64 fragment reads.
// EPI 0: C row-major [M x NN]. EPI 1: heads layout
// C[((b*8+h)*1024+pos)*64+dd] with row=b*1024+pos, col=h*64+dd.
#define KC 32
#define LROW 36
template <int KDIM, int EPI>
__global__ __launch_bounds__(128) void gemm_async_kernel(
    const float* __restrict__ A, const float* __restrict__ W,
    const float* __restrict__ bias, float* __restrict__ C, int NN) {
  __shared__ __align__(16) float sA[2][32 * LROW];
  __shared__ __align__(16) float sW[2][32 * LROW];
  int mBase = blockIdx.x * 32;
  int nBase = blockIdx.y * 32;
  int tid = threadIdx.x;
  int wave = tid >> 5, lane = tid & 31;
  int wm = wave >> 1, wn = wave & 1;
  int r = lane & 15, half = lane >> 4;

  // stage one 32x32-float chunk of A and of W into LDS buffer `buf`
  auto stage = [&](int kc, int buf) {
    for (int c = tid; c < 256; c += 128) {  // 256 16B-chunks per matrix
      int row = c >> 3, col = (c & 7) * 4;
      async_copy16(&sA[buf][row * LROW + col],
                   A + (size_t)(mBase + row) * KDIM + kc + col);
      async_copy16(&sW[buf][row * LROW + col],
                   W + (size_t)(nBase + row) * KDIM + kc + col);
    }
  };

  v8f acc = {};
  stage(0, 0);
  const int NSTEP = KDIM / KC;
  for (int s = 0; s < NSTEP; ++s) {
    int buf = s & 1;
    wait_async0();
    __syncthreads();  // chunk s resident for all waves
    if (s + 1 < NSTEP) stage((s + 1) * KC, buf ^ 1);  // prefetch next
    const float* a0 = &sA[buf][(wm * 16 + r) * LROW + 2 * half];
    const float* w0 = &sW[buf][(wn * 16 + r) * LROW + 2 * half];
#pragma unroll
    for (int k = 0; k < KC; k += 4) {
      v2f a = *(const v2f*)(a0 + k);
      v2f b = *(const v2f*)(w0 + k);
      acc = wmma4(a, b, acc);
    }
    __syncthreads();  // all reads of buf done before it is restaged
  }

  int m0 = mBase + wm * 16, n0 = nBase + wn * 16;
  float bn = bias[n0 + r];
#pragma unroll
  for (int vg = 0; vg < 8; ++vg) {
    int row = m0 + vg + 8 * half;
    int col = n0 + r;
    float v = acc[vg] + bn;
    if (EPI == 0) {
      C[(size_t)row * NN + col] = v;
    } else {
      int b = row >> 10, pos = row & 1023;
      int h = col >> 6, dd = col & 63;
      C[(((size_t)b * HEADS + h) * NPOS + pos) * DHEAD + dd] = v;
    }
  }
}

// ----------------------------------------------------------- attention ----
// per block: one (b,h), one 16-row tile. S[16][1024] in LDS (stride 1028 to
// spread banks). QK^T and P@V both via fp32 WMMA; softmax wave-cooperative.
#define SROW 1028
__global__ void attention_kernel(const float* __restrict__ Q,
                                 const float* __restrict__ K,
                                 const float* __restrict__ V,
                                 const int* __restrict__ midx,
                                 float* __restrict__ AO) {
  __shared__ float S[16 * SROW];
  __shared__ int mrow[16][4];
  int bh = blockIdx.x;       // 0..31
  int m0 = blockIdx.y * 16;  // row tile within (b,h)
  int tid = threadIdx.x;     // 0..127
  int wave = tid >> 5, lane = tid & 31;
  int r = lane & 15, half = lane >> 4;
  const float* Qb = Q + (size_t)bh * NPOS * DHEAD;
  const float* Kb = K + (size_t)bh * NPOS * DHEAD;
  const float* Vb = V + (size_t)bh * NPOS * DHEAD;
  if (tid < 64)
    mrow[tid >> 2][tid & 3] =
        midx[((size_t)bh * NPOS + m0 + (tid >> 2)) * 4 + (tid & 3)];
  __syncthreads();

  // S = (Q K^T) * scale * hard_mask
  const float* qrow = Qb + (size_t)(m0 + r) * DHEAD + 2 * half;
  for (int jt = wave; jt < 64; jt += 4) {
    int j0 = jt * 16;
    const float* krow = Kb + (size_t)(j0 + r) * DHEAD + 2 * half;
    v8f acc = {};
#pragma unroll
    for (int k = 0; k < DHEAD; k += 4) {
      v2f a = *(const v2f*)(qrow + k);
      v2f b = *(const v2f*)(krow + k);
      acc = wmma4(a, b, acc);
    }
    int j = j0 + r;
#pragma unroll
    for (int vg = 0; vg < 8; ++vg) {
      int row = vg + 8 * half;
      bool hit = (j == mrow[row][0]) | (j == mrow[row][1]) |
                 (j == mrow[row][2]) | (j == mrow[row][3]);
      S[row * SROW + j] = acc[vg] * SCALE * (hit ? 1.0f : -100000.0f);
    }
  }
  __syncthreads();

  // row softmax (wave handles 4 rows)
  for (int rr = 0; rr < 4; ++rr) {
    int row = wave * 4 + rr;
    float* Sr = S + row * SROW;
    float mx = -INFINITY;
    for (int j = lane; j < NPOS; j += 32) mx = fmaxf(mx, Sr[j]);
#pragma unroll
    for (int off = 16; off >= 1; off >>= 1) mx = fmaxf(mx, __shfl_xor(mx, off));
    float sum = 0.f;
    for (int j = lane; j < NPOS; j += 32) {
      float e = __expf(Sr[j] - mx);
      Sr[j] = e;
      sum += e;
    }
#pragma unroll
    for (int off = 16; off >= 1; off >>= 1) sum += __shfl_xor(sum, off);
    float inv = 1.0f / sum;
    for (int j = lane; j < NPOS; j += 32) Sr[j] *= inv;
  }
  __syncthreads();

  // O = P @ V ; wave owns d-tile n0 = wave*16; A-frags come from LDS
  {
    int n0 = wave * 16;
    v8f acc = {};
    const float* sa = S + r * SROW + 2 * half;
#pragma unroll 4
    for (int k = 0; k < NPOS; k += 4) {
      v2f a;
      a.x = sa[k];
      a.y = sa[k + 1];
      const float* vp = Vb + (size_t)(k + 2 * half) * DHEAD + n0 + r;
      v2f b;
      b.x = vp[0];
      b.y = vp[DHEAD];
      acc = wmma4(a, b, acc);
    }
    int b_ = bh >> 3, h_ = bh & 7;
    int col = h_ * DHEAD + n0 + r;
#pragma unroll
    for (int vg = 0; vg < 8; ++vg) {
      int pos = m0 + vg + 8 * half;
      AO[((size_t)b_ * NPOS + pos) * INNER + col] = acc[vg];
    }
  }
}

// ------------------------------------------------------------- launcher ---
extern "C" void kernel_launch(void* const* d_in, const int* in_sizes, int n_in,
                              void* d_out, int out_size, void* d_ws,
                              size_t ws_size, hipStream_t stream) {
  auto in = [&](int i) { return (const float*)d_in[i]; };
  const float* x = in(0);
  // input order: x, then per p in {q,k,v}: pd_w, pd_b, p_g, p_beta, pp_w,
  // pp_b, pm_w, pm_b; then out_w, out_b.
  char* ws = (char*)d_ws;
  size_t off = 0;
  auto alloc = [&](size_t bytes) {
    void* p = ws + off;
    off += (bytes + 255) & ~(size_t)255;
    return p;
  };
  float* xattn = (float*)alloc((size_t)BATCH * 2 * NPOS * 4);
  float* qm = (float*)alloc((size_t)BATCH * HEADS * NPOS * 2 * 4);
  float* km = (float*)alloc((size_t)BATCH * HEADS * NPOS * 2 * 4);
  int* midx = (int*)alloc((size_t)BATCH * HEADS * NPOS * 4 * 4);
  float* act = (float*)alloc((size_t)MROWS * DIM * 4);
  float* qkv[3];
  for (int i = 0; i < 3; ++i)
    qkv[i] = (float*)alloc((size_t)BATCH * HEADS * NPOS * DHEAD * 4);
  float* ao = (float*)alloc((size_t)MROWS * INNER * 4);

  pool_kernel<<<MROWS / 256, 256, 0, stream>>>(x, xattn);
  maskproj_kernel<<<(BATCH * 16 * NPOS * 2) / 256, 256, 0, stream>>>(
      xattn, in(7), in(8), in(15), in(16), qm, km);
  topk_kernel<<<BATCH * HEADS * NPOS / 4, 128, 0, stream>>>(qm, km, midx);
  for (int p = 0; p < 3; ++p) {
    int base = 1 + p * 8;
    sepconv_kernel<<<DIM, 256, 0, stream>>>(x, in(base + 0), in(base + 1),
                                            in(base + 2), in(base + 3), act);
    gemm_async_kernel<DIM, 1><<<dim3(MROWS / 32, INNER / 32), 128, 0, stream>>>(
        act, in(base + 4), in(base + 5), qkv[p], INNER);
  }
  attention_kernel<<<dim3(BATCH * HEADS, NPOS / 16), 128, 0, stream>>>(
      qkv[0], qkv[1], qkv[2], midx, ao);
  gemm_async_kernel<INNER, 0><<<dim3(MROWS / 32, DIM / 32), 128, 0, stream>>>(
      ao, in(25), in(26), (float*)d_out, DIM);
}